// YoloLoss_36481452212371
// MI455X (gfx1250) — compile-verified
//
#include <hip/hip_runtime.h>

typedef __attribute__((ext_vector_type(2))) float v2f;
typedef __attribute__((ext_vector_type(8))) float v8f;

#define CELL_D 30          // 2 boxes * 5 + 20 classes
#define CELLS_PER_IMG 49   // 7*7
#define LCOORD 5.0f
#define LNOOBJ 0.5f

__device__ __forceinline__ float cell_contrib(const float* __restrict__ pred,
                                              const float* __restrict__ tgt,
                                              int cell)
{
    float pv[CELL_D], tv[CELL_D];
    const float2* p2 = reinterpret_cast<const float2*>(pred + (size_t)cell * CELL_D);
    const float2* t2 = reinterpret_cast<const float2*>(tgt  + (size_t)cell * CELL_D);
#pragma unroll
    for (int i = 0; i < CELL_D / 2; ++i) {
        float2 a = p2[i]; pv[2*i] = a.x; pv[2*i+1] = a.y;
        float2 b = t2[i]; tv[2*i] = b.x; tv[2*i+1] = b.y;
    }

    // ---- target box 0 in xyxy (xy scaled by 1/7) ----
    float tcx = tv[0] / 7.0f, tcy = tv[1] / 7.0f;
    float t_x1 = tcx - 0.5f * tv[2], t_x2 = tcx + 0.5f * tv[2];
    float t_y1 = tcy - 0.5f * tv[3], t_y2 = tcy + 0.5f * tv[3];
    float t_area = (t_x2 - t_x1) * (t_y2 - t_y1);

    // ---- IoU of each predicted box vs target box 0 ----
    float iou0 = 0.0f, iou1 = 0.0f;
#pragma unroll
    for (int b = 0; b < 2; ++b) {
        float bx = (b == 0) ? pv[0] : pv[5];
        float by = (b == 0) ? pv[1] : pv[6];
        float bw = (b == 0) ? pv[2] : pv[7];
        float bh = (b == 0) ? pv[3] : pv[8];
        float cx = bx / 7.0f, cy = by / 7.0f;
        float x1 = cx - 0.5f * bw, x2 = cx + 0.5f * bw;
        float y1 = cy - 0.5f * bh, y2 = cy + 0.5f * bh;
        float iw = fmaxf(fminf(x2, t_x2) - fmaxf(x1, t_x1), 0.0f);
        float ih = fmaxf(fminf(y2, t_y2) - fmaxf(y1, t_y1), 0.0f);
        float inter = iw * ih;
        float area  = (x2 - x1) * (y2 - y1);
        float iou   = inter / (area + t_area - inter + 1e-10f);
        if (b == 0) iou0 = iou; else iou1 = iou;
    }
    bool  r1      = iou1 > iou0;        // argmax, ties -> box 0
    float max_iou = fmaxf(iou0, iou1);

    // ---- responsible box (branchless selects, keeps arrays in VGPRs) ----
    float prx = r1 ? pv[5] : pv[0];
    float pry = r1 ? pv[6] : pv[1];
    float prw = r1 ? pv[7] : pv[2];
    float prh = r1 ? pv[8] : pv[3];
    float prc = r1 ? pv[9] : pv[4];
    float trx = r1 ? tv[5] : tv[0];
    float try_ = r1 ? tv[6] : tv[1];
    float trw = r1 ? tv[7] : tv[2];
    float trh = r1 ? tv[8] : tv[3];

    float dx = prx - trx, dy = pry - try_;
    float loss_xy = dx * dx + dy * dy;
    float sw = __builtin_sqrtf(prw) - __builtin_sqrtf(trw);
    float sh = __builtin_sqrtf(prh) - __builtin_sqrtf(trh);
    float loss_wh = sw * sw + sh * sh;
    float dc = prc - max_iou;
    float loss_obj = dc * dc;

    float loss_cls = 0.0f;
#pragma unroll
    for (int c = 10; c < CELL_D; ++c) {
        float d = pv[c] - tv[c];
        loss_cls = __builtin_fmaf(d, d, loss_cls);
    }

    float d0 = pv[4] - tv[4], d1 = pv[9] - tv[9];
    float loss_noobj = d0 * d0 + d1 * d1;

    bool obj = tv[4] > 0.0f;    // m_noobj == (tv[4] == 0) exactly complements
    return obj ? (LCOORD * (loss_xy + loss_wh) + loss_obj + loss_cls)
               : (LNOOBJ * loss_noobj);
}

__global__ void __launch_bounds__(256)
yolo_partial_kernel(const float* __restrict__ pred,
                    const float* __restrict__ tgt,
                    float* __restrict__ partial,
                    int ncells)
{
    __shared__ float wsum[8];
    float acc = 0.0f;
    int stride = gridDim.x * blockDim.x;
    for (int cell = blockIdx.x * blockDim.x + threadIdx.x; cell < ncells; cell += stride)
        acc += cell_contrib(pred, tgt, cell);

    // ---- wave32 reduction via V_WMMA_F32_16X16X4_F32 ----
    // A = ones(16x4), B = [acc row; zeros] => D[m][n] = acc[n] + acc[n+16]
    // (EXEC is all-ones here: loop has reconverged, as WMMA requires.)
    v2f a; a.x = 1.0f; a.y = 1.0f;
    v2f b; b.x = acc;  b.y = 0.0f;
    v8f c = {};
    c = __builtin_amdgcn_wmma_f32_16x16x4_f32(
            /*neg_a=*/false, a, /*neg_b=*/false, b,
            /*c_mod=*/(short)0, c, /*reuse_a=*/false, /*reuse_b=*/false);
    float r = c[0];                     // lane L: acc[L%16] + acc[L%16+16]
    r += __shfl_xor(r, 1, 32);
    r += __shfl_xor(r, 2, 32);
    r += __shfl_xor(r, 4, 32);
    r += __shfl_xor(r, 8, 32);          // all lanes: wave total

    int lane = threadIdx.x & 31;
    int wave = threadIdx.x >> 5;
    if (lane == 0) wsum[wave] = r;
    __syncthreads();
    if (threadIdx.x == 0) {
        float s = 0.0f;
        int nw = blockDim.x >> 5;
        for (int w = 0; w < nw; ++w) s += wsum[w];   // fixed order: deterministic
        partial[blockIdx.x] = s;
    }
}

__global__ void __launch_bounds__(256)
yolo_final_kernel(const float* __restrict__ partial, int nparts,
                  float* __restrict__ out, float inv_n)
{
    __shared__ float sh[256];
    float s = 0.0f;
    for (int i = threadIdx.x; i < nparts; i += 256) s += partial[i];
    sh[threadIdx.x] = s;
    __syncthreads();
#pragma unroll
    for (int off = 128; off > 0; off >>= 1) {
        if (threadIdx.x < off) sh[threadIdx.x] += sh[threadIdx.x + off];
        __syncthreads();
    }
    if (threadIdx.x == 0) out[0] = sh[0] * inv_n;
}

extern "C" void kernel_launch(void* const* d_in, const int* in_sizes, int n_in,
                              void* d_out, int out_size, void* d_ws, size_t ws_size,
                              hipStream_t stream)
{
    const float* pred = (const float*)d_in[0];
    const float* tgt  = (const float*)d_in[1];
    float* out        = (float*)d_out;
    float* partial    = (float*)d_ws;

    long long total = (long long)in_sizes[0];       // N*7*7*30
    int ncells = (int)(total / CELL_D);             // N*49
    float inv_n = (float)CELLS_PER_IMG / (float)ncells;  // 1/N

    int G = 1024;
    if (ws_size < (size_t)G * sizeof(float)) G = (int)(ws_size / sizeof(float));
    if (G < 1) G = 1;

    yolo_partial_kernel<<<G, 256, 0, stream>>>(pred, tgt, partial, ncells);
    yolo_final_kernel<<<1, 256, 0, stream>>>(partial, G, out, inv_n);
}